// stabNet_7473243095657
// MI455X (gfx1250) — compile-verified
//
#include <hip/hip_runtime.h>
#include <hip/hip_bf16.h>
#include <math.h>

#define GH 4
#define GW 4
#define H_IMG 720
#define W_IMG 1280
#define BATCH 8
#define CHN 3

typedef __attribute__((ext_vector_type(2))) float v2f;
typedef __attribute__((ext_vector_type(8))) float v8f;

// ---------------------------------------------------------------------------
// Stage 1: 128 independent 8x8 solves (one per (b, cell_i, cell_j)).
// Builds the DLT system for the 4 cell corners -> target points, adds 1e-4*I,
// LU with partial pivoting (matches jnp.linalg.solve), emits 3x3 H (h8=1).
// ---------------------------------------------------------------------------
__global__ void stab_solve_kernel(const float* __restrict__ theta,
                                  float* __restrict__ Hs) {
    int t = blockIdx.x * blockDim.x + threadIdx.x;
    if (t >= BATCH * GH * GW) return;
    int b  = t / (GH * GW);
    int r  = t - b * (GH * GW);
    int ci = r / GW;
    int cj = r - ci * GW;

    const float hc = 2.0f / GH;
    const float wc = 2.0f / GW;
    float hh = ci * hc - 1.0f;
    float ww = cj * wc - 1.0f;

    float xc[4] = { ww, ww + wc, ww, ww + wc };
    float yc[4] = { hh, hh,      hh + hc, hh + hc };

    // theta: [B, GH+1, GW+1, 2]; corners t00,t01,t10,t11
    auto T = [&](int i, int j, int k) {
        return theta[(((b * (GH + 1)) + i) * (GW + 1) + j) * 2 + k];
    };
    float u[4] = { T(ci, cj, 0), T(ci, cj + 1, 0), T(ci + 1, cj, 0), T(ci + 1, cj + 1, 0) };
    float v[4] = { T(ci, cj, 1), T(ci, cj + 1, 1), T(ci + 1, cj, 1), T(ci + 1, cj + 1, 1) };

    float M[8][9];
    for (int p = 0; p < 4; ++p) {
        M[p][0] = xc[p]; M[p][1] = yc[p]; M[p][2] = 1.0f;
        M[p][3] = 0.0f;  M[p][4] = 0.0f;  M[p][5] = 0.0f;
        M[p][6] = -xc[p] * u[p]; M[p][7] = -yc[p] * u[p];
        M[p][8] = u[p];
        M[p + 4][0] = 0.0f;  M[p + 4][1] = 0.0f;  M[p + 4][2] = 0.0f;
        M[p + 4][3] = xc[p]; M[p + 4][4] = yc[p]; M[p + 4][5] = 1.0f;
        M[p + 4][6] = -xc[p] * v[p]; M[p + 4][7] = -yc[p] * v[p];
        M[p + 4][8] = v[p];
    }
    for (int i = 0; i < 8; ++i) M[i][i] += 1.0e-4f;

    // LU with partial pivoting, forward elimination on augmented matrix
    for (int col = 0; col < 8; ++col) {
        int piv = col;
        float best = fabsf(M[col][col]);
        for (int rr = col + 1; rr < 8; ++rr) {
            float av = fabsf(M[rr][col]);
            if (av > best) { best = av; piv = rr; }
        }
        if (piv != col) {
            for (int k = col; k < 9; ++k) {
                float tmp = M[col][k]; M[col][k] = M[piv][k]; M[piv][k] = tmp;
            }
        }
        float inv = 1.0f / M[col][col];
        for (int rr = col + 1; rr < 8; ++rr) {
            float f = M[rr][col] * inv;
            for (int k = col; k < 9; ++k) M[rr][k] -= f * M[col][k];
        }
    }
    float sol[8];
    for (int rr = 7; rr >= 0; --rr) {
        float s = M[rr][8];
        for (int k = rr + 1; k < 8; ++k) s -= M[rr][k] * sol[k];
        sol[rr] = s / M[rr][rr];
    }
    float* hp = Hs + t * 9;
    for (int k = 0; k < 8; ++k) hp[k] = sol[k];
    hp[8] = 1.0f;
}

// ---------------------------------------------------------------------------
// Stage 2: per-pixel projective warp + bilinear gather.
// Each wave handles 32 consecutive x in one row (320-wide cells => one cell
// per wave, uniform H). The 3x3 map is done on the matrix pipe:
//   A(16x4) rows {0,1,2, 8,9,10} = H rows, B(4x16) cols = (gx, gy, 1, 0).
// Two v_wmma_f32_16x16x4_f32 (pixels 0-15 and 16-31); per the C/D layout
// lanes 0-15 read c0[0..2] (M=0..2) and lanes 16-31 read c1[0..2] (M=8..10),
// so every lane ends up with its own pixel's (xs, ys, zs) w/o shuffles.
// No divergence: EXEC stays all-ones as WMMA requires.
// ---------------------------------------------------------------------------
__global__ __launch_bounds__(256) void stab_warp_kernel(
    const float* __restrict__ U,      // [B, C, H, W]
    const float* __restrict__ Hs,     // [B, GH, GW, 9]
    float* __restrict__ out_img,      // [B, C, H, W]
    float* __restrict__ out_black,    // [B, H, W]
    float* __restrict__ out_map,      // [B, H, W, 2]
    float* __restrict__ out_x,        // [B, H, W, 1]
    float* __restrict__ out_y) {      // [B, H, W, 1]

    const int W = W_IMG, H = H_IMG;
    int x = blockIdx.x * 256 + threadIdx.x;
    int y = blockIdx.y;
    int b = blockIdx.z;
    int lane = threadIdx.x & 31;
    int l15  = lane & 15;
    bool lowK = lane < 16;

    int ci = y / (H / GH);            // 180 rows per cell
    int cj = x / (W / GW);            // 320 cols per cell; uniform per wave
    const float* hp = Hs + ((b * GH + ci) * GW + cj) * 9;

    // ---- A matrix: rows 0-2 and 8-10 = H rows (16x4 f32 layout:
    // lanes 0-15: VGPR0=K0, VGPR1=K1; lanes 16-31: VGPR0=K2, VGPR1=K3)
    int mrow = lane & 7;                         // valid lanes carry row mrow
    int mcl  = mrow < 2 ? mrow : 2;              // clamp for safe load
    float h0 = hp[3 * mcl + (lowK ? 0 : 2)];
    float h1 = hp[3 * mcl + 1];
    bool valid = mrow < 3;                       // lanes 0-2,8-10,16-18,24-26
    v2f A;
    A.x = valid ? h0 : 0.0f;
    A.y = (valid && lowK) ? h1 : 0.0f;

    // ---- B matrices: col n = (gx_n, gy, 1, 0); two groups of 16 pixels
    int xbase = x - lane;
    float gy  = -1.0f + 2.0f * (float)y / (float)(H - 1);
    float gx0 = -1.0f + 2.0f * (float)(xbase + l15) / (float)(W - 1);
    float gx1 = -1.0f + 2.0f * (float)(xbase + 16 + l15) / (float)(W - 1);
    v2f B0, B1;
    B0.x = lowK ? gx0 : 1.0f;  B0.y = lowK ? gy : 0.0f;
    B1.x = lowK ? gx1 : 1.0f;  B1.y = lowK ? gy : 0.0f;

    v8f cz = {};
    v8f c0 = __builtin_amdgcn_wmma_f32_16x16x4_f32(
        false, A, false, B0, (short)0, cz, false, false);
    v8f c1 = __builtin_amdgcn_wmma_f32_16x16x4_f32(
        false, A, false, B1, (short)0, cz, false, false);

    float xs = lowK ? c0[0] : c1[0];
    float ys = lowK ? c0[1] : c1[1];
    float zs = lowK ? c0[2] : c1[2];

    // ---- perspective divide with signed epsilon (matches reference)
    float sz = (zs >= 0.0f) ? 1.0f : -1.0f;
    zs += sz * 1.0e-8f;
    float xm = xs / zs;
    float ym = ys / zs;

    bool oob = (xm < -1.0f) || (xm > 1.0f) || (ym < -1.0f) || (ym > 1.0f);

    // ---- bilinear gather (reference clips indices, weights use clipped idx)
    float xf = (xm + 1.0f) * ((float)W * 0.5f);
    float yf = (ym + 1.0f) * ((float)H * 0.5f);
    float xff = floorf(fminf(fmaxf(xf, -1.0e9f), 1.0e9f));
    float yff = floorf(fminf(fmaxf(yf, -1.0e9f), 1.0e9f));
    int x0 = (int)xff, y0 = (int)yff;
    int x1 = x0 + 1, y1 = y0 + 1;
    x0 = min(max(x0, 0), W - 1);
    x1 = min(max(x1, 0), W - 1);
    y0 = min(max(y0, 0), H - 1);
    y1 = min(max(y1, 0), H - 1);
    float x0f = (float)x0, x1f = (float)x1;
    float y0f = (float)y0, y1f = (float)y1;
    float wa = (x1f - xf) * (y1f - yf);
    float wb = (x1f - xf) * (yf - y0f);
    float wdc = (xf - x0f) * (y1f - yf);
    float wd = (xf - x0f) * (yf - y0f);

    int pix = (b * H + y) * W + x;
#pragma unroll
    for (int c = 0; c < CHN; ++c) {
        const float* ub = U + ((b * CHN + c) * H) * W;
        float Ia = ub[y0 * W + x0];
        float Ib = ub[y1 * W + x0];
        float Ic = ub[y0 * W + x1];
        float Id = ub[y1 * W + x1];
        float val = wa * Ia + wb * Ib + wdc * Ic + wd * Id;
        out_img[((b * CHN + c) * H + y) * W + x] = val;
    }
    out_black[pix]       = oob ? 1.0f : 0.0f;
    out_map[pix * 2 + 0] = xm;
    out_map[pix * 2 + 1] = ym;
    out_x[pix]           = xm;
    out_y[pix]           = ym;
}

extern "C" void kernel_launch(void* const* d_in, const int* in_sizes, int n_in,
                              void* d_out, int out_size, void* d_ws, size_t ws_size,
                              hipStream_t stream) {
    const float* U     = (const float*)d_in[0];   // [8,3,720,1280] f32
    const float* theta = (const float*)d_in[1];   // [8,5,5,2]      f32

    float* Hs = (float*)d_ws;                     // 128*9 floats of scratch

    float* out = (float*)d_out;
    const int BHW = BATCH * H_IMG * W_IMG;
    float* out_img   = out;                       // [B,C,H,W]
    float* out_black = out_img + (size_t)CHN * BHW;
    float* out_map   = out_black + BHW;           // [B,H,W,2]
    float* out_x     = out_map + 2 * (size_t)BHW;
    float* out_y     = out_x + BHW;

    stab_solve_kernel<<<1, 128, 0, stream>>>(theta, Hs);

    dim3 grid(W_IMG / 256, H_IMG, BATCH);         // 5 x 720 x 8, exact tiling
    stab_warp_kernel<<<grid, 256, 0, stream>>>(U, Hs, out_img, out_black,
                                               out_map, out_x, out_y);
}